// RoutingMoudule_5257039970818
// MI455X (gfx1250) — compile-verified
//
#include <hip/hip_runtime.h>

typedef __attribute__((ext_vector_type(2))) float v2f;
typedef __attribute__((ext_vector_type(8))) float v8f;

#define BATCH   128
#define IN_CAPS 1152
#define IN_DIM  8
#define OUT_CAPS 10
#define OUT_DIM  16
#define NI      4   // capsules per wave in the WMMA kernel

// -----------------------------------------------------------------------------
// Kernel A: u_hat[b][o][i][q] = sum_p W[o][i][q][p] * x[b][i][p]
// One wave per (b_tile of 16, o, 4 consecutive i). M=batch(16), N=q(16),
// K=p(8) as two chained V_WMMA_F32_16X16X4_F32 (fp32, no precision loss).
//
// A operand layout (16x4 f32): lane l -> M = l&15, VGPR r holds K = r + 2*(l>>4)
// B operand layout (4x16 f32): lane l -> N = l&15, VGPR r holds K = r + 2*(l>>4)
// C/D layout (16x16 f32):      lane l -> N = l&15, VGPR r holds M = r + 8*(l>>4)
//
// All per-capsule / per-row offsets are compile-time constants that fit the
// signed 24-bit IOFFSET (max store offset 7*737280 B = 5.16 MB), so each wave
// uses just three base address pairs.
// -----------------------------------------------------------------------------
__global__ __launch_bounds__(256) void uhat_wmma_kernel(
    const float* __restrict__ x,     // [B][IC][ID]
    const float* __restrict__ W,     // [OC][IC][OD][ID]
    float* __restrict__ uhat)        // [B][OC][IC][OD]
{
  const int wave = blockIdx.x * (blockDim.x >> 5) + (threadIdx.x >> 5);
  const int lane = threadIdx.x & 31;

  const int ic = wave % (IN_CAPS / NI);        // capsule chunk (0..287)
  const int t  = wave / (IN_CAPS / NI);
  const int o  = t % OUT_CAPS;
  const int b0 = (t / OUT_CAPS) * 16;          // batch tile base (0..112)
  const int i0 = ic * NI;

  const int m  = lane & 15;                    // A: batch row / B: q col / D: N
  const int kh = lane >> 4;                    // K-half selector

  const float* xp = x + ((size_t)(b0 + m) * IN_CAPS + i0) * IN_DIM + 2 * kh;
  const float* wp = W + (((size_t)o * IN_CAPS + i0) * OUT_DIM + m) * IN_DIM + 2 * kh;
  float*       up = uhat + (((size_t)(b0 + 8 * kh) * OUT_CAPS + o) * IN_CAPS + i0) * OUT_DIM + m;

  const int MSTRIDE = OUT_CAPS * IN_CAPS * OUT_DIM;   // elements between batch rows

#pragma unroll
  for (int ii = 0; ii < NI; ++ii) {
    v8f acc = {};
#pragma unroll
    for (int ks = 0; ks < 2; ++ks) {
      const v2f a    = *(const v2f*)(xp + (size_t)ii * IN_DIM + ks * 4);
      const v2f bmat = *(const v2f*)(wp + (size_t)ii * OUT_DIM * IN_DIM + ks * 4);
      acc = __builtin_amdgcn_wmma_f32_16x16x4_f32(
          /*neg_a=*/false, a, /*neg_b=*/false, bmat,
          /*c_mod=*/(short)0, acc, /*reuse_a=*/false, /*reuse_b=*/false);
    }
#pragma unroll
    for (int r = 0; r < 8; ++r)
      up[(size_t)ii * OUT_DIM + (size_t)r * MSTRIDE] = acc[r];
  }
}

// -----------------------------------------------------------------------------
// Kernel B: dynamic routing, one workgroup per batch element.
// LDS: logits b[10][1152] + per-i softmax stats; u_hat rows streamed from L2.
// Outputs (v, v1, v2) concatenated flat in d_out.
// -----------------------------------------------------------------------------
#define RT_THREADS 320

__global__ __launch_bounds__(RT_THREADS) void routing_kernel(
    const float* __restrict__ uhat,  // [B][OC][IC][OD]
    float* __restrict__ out)         // v | v1 | v2, each [B][OC][OD]
{
  const int b   = blockIdx.x;
  const int tid = threadIdx.x;

  __shared__ float blog[OUT_CAPS * IN_CAPS];           // routing logits
  __shared__ float mx[IN_CAPS];                        // softmax max over o
  __shared__ float dn[IN_CAPS];                        // softmax denom over o
  __shared__ float s_lds[OUT_CAPS][OUT_DIM];
  __shared__ float part[2][OUT_CAPS][OUT_DIM];
  __shared__ float pnorm[OUT_CAPS];

  const float* U = uhat + (size_t)b * OUT_CAPS * IN_CAPS * OUT_DIM;
  const int BOD = BATCH * OUT_CAPS * OUT_DIM;

  for (int idx = tid; idx < OUT_CAPS * IN_CAPS; idx += RT_THREADS) blog[idx] = 0.f;
  __syncthreads();

  for (int itr = 0; itr < 3; ++itr) {
    // ---- Phase A: softmax stats over o, per i ----
    for (int i = tid; i < IN_CAPS; i += RT_THREADS) {
      float m = blog[i];
      for (int o = 1; o < OUT_CAPS; ++o) m = fmaxf(m, blog[o * IN_CAPS + i]);
      float d = 0.f;
      for (int o = 0; o < OUT_CAPS; ++o) d += __expf(blog[o * IN_CAPS + i] - m);
      mx[i] = m; dn[i] = d;
    }
    __syncthreads();

    // ---- Phase B: s[o][d] = sum_i softmax(b)[o][i] * u_hat[o][i][d] ----
    {
      const int p    = tid % 160;
      const int half = tid / 160;               // 0 or 1
      const int o = p / OUT_DIM, d = p % OUT_DIM;
      const int i0 = half * (IN_CAPS / 2), i1 = i0 + IN_CAPS / 2;
      float acc = 0.f;
      for (int i = i0; i < i1; ++i) {
        const float c = __expf(blog[o * IN_CAPS + i] - mx[i]) / dn[i];
        acc = fmaf(c, U[((size_t)o * IN_CAPS + i) * OUT_DIM + d], acc);
      }
      part[half][o][d] = acc;
    }
    __syncthreads();
    if (tid < 160) {
      const int o = tid / OUT_DIM, d = tid % OUT_DIM;
      s_lds[o][d] = part[0][o][d] + part[1][o][d];
    }
    __syncthreads();

    if (itr < 2) {
      // ---- Phase C: v1/v2 = squash(s);  ||squash(s)|| = sq/(1+sq) ----
      if (tid < OUT_CAPS) {
        const int o = tid;
        float sq = 0.f;
        for (int d = 0; d < OUT_DIM; ++d) sq += s_lds[o][d] * s_lds[o][d];
        const float f = sq / ((1.f + sq) * sqrtf(sq));
        float* dst = out + (size_t)(itr + 1) * BOD + ((size_t)b * OUT_CAPS + o) * OUT_DIM;
        for (int d = 0; d < OUT_DIM; ++d) dst[d] = f * s_lds[o][d];
        if (itr == 1) pnorm[o] = sq / (1.f + sq);
      }
      __syncthreads();

      // ---- Phase D: b += pearson(u_hat_row, squash(s - c*u_hat_row)) ----
      // Pearson is invariant to the positive squash scale, so skip it.
      for (int idx = tid; idx < OUT_CAPS * IN_CAPS; idx += RT_THREADS) {
        const int o = idx / IN_CAPS, i = idx % IN_CAPS;
        const float* Ur = U + ((size_t)o * IN_CAPS + i) * OUT_DIM;
        float tr[OUT_DIM], vr[OUT_DIM];
#pragma unroll
        for (int d = 0; d < OUT_DIM; ++d) tr[d] = Ur[d];
        const float c = __expf(blog[o * IN_CAPS + i] - mx[i]) / dn[i];
#pragma unroll
        for (int d = 0; d < OUT_DIM; ++d) vr[d] = s_lds[o][d] - c * tr[d];
        float am = 0.f, vm = 0.f;
#pragma unroll
        for (int d = 0; d < OUT_DIM; ++d) { am += tr[d]; vm += vr[d]; }
        am *= (1.f / OUT_DIM); vm *= (1.f / OUT_DIM);
        float num = 0.f, na = 0.f, nb = 0.f;
#pragma unroll
        for (int d = 0; d < OUT_DIM; ++d) {
          const float ta = tr[d] - am, vb = vr[d] - vm;
          num = fmaf(ta, vb, num); na = fmaf(ta, ta, na); nb = fmaf(vb, vb, nb);
        }
        blog[o * IN_CAPS + i] += num / (sqrtf(na) * sqrtf(nb));
      }
      __syncthreads();
    } else {
      // ---- Final: v = squash(pnorm[o] * s) ----
      if (tid < OUT_CAPS) {
        const int o = tid;
        float sv[OUT_DIM];
        float sq = 0.f;
        for (int d = 0; d < OUT_DIM; ++d) {
          sv[d] = pnorm[o] * s_lds[o][d];
          sq += sv[d] * sv[d];
        }
        const float f = sq / ((1.f + sq) * sqrtf(sq));
        float* dst = out + ((size_t)b * OUT_CAPS + o) * OUT_DIM;
        for (int d = 0; d < OUT_DIM; ++d) dst[d] = f * sv[d];
      }
    }
  }
}

extern "C" void kernel_launch(void* const* d_in, const int* in_sizes, int n_in,
                              void* d_out, int out_size, void* d_ws, size_t ws_size,
                              hipStream_t stream) {
  const float* x = (const float*)d_in[0];   // [128,1152,8]
  const float* W = (const float*)d_in[1];   // [1,10,1152,16,8]
  float* out  = (float*)d_out;              // v|v1|v2, 3*128*10*16 floats
  float* uhat = (float*)d_ws;               // [128][10][1152][16] = 94.4 MB

  // 8 batch-tiles * 10 out-caps * 288 capsule-chunks = 23040 waves, 8/block
  const int total_waves = (BATCH / 16) * OUT_CAPS * (IN_CAPS / NI);
  uhat_wmma_kernel<<<total_waves / 8, 256, 0, stream>>>(x, W, uhat);
  routing_kernel<<<BATCH, RT_THREADS, 0, stream>>>(uhat, out);
}